// SingleBlock_36197984371014
// MI455X (gfx1250) — compile-verified
//
#include <hip/hip_runtime.h>
#include <math.h>

// ---------------------------------------------------------------------------
// CDNA5 (gfx1250) fp32 WMMA implementation of the 2-block dual-attention net.
// All GEMMs and both attention einsums run on V_WMMA_F32_16X16X4_F32 (wave32).
// Global->LDS staging uses GLOBAL_LOAD_ASYNC_TO_LDS_B128 (ASYNCcnt) with
// double-buffered LDS tiles so the async DMA overlaps the WMMA work.
// ---------------------------------------------------------------------------

typedef float v2f __attribute__((ext_vector_type(2)));
typedef float v8f __attribute__((ext_vector_type(8)));

#define O_DIM 1024

#if defined(__AMDGCN__) &&                                             \
    __has_builtin(__builtin_amdgcn_global_load_async_to_lds_b128) &&    \
    __has_builtin(__builtin_amdgcn_s_wait_asynccnt)
#define HAVE_ASYNC_LDS 1
#else
#define HAVE_ASYNC_LDS 0
#endif

#if HAVE_ASYNC_LDS
// builtin prototype (from compiler diagnostic): param0 = int4 addrspace(1)*,
// param1 = int4 addrspace(3)*, param2 = imm offset, param3 = imm cpol
typedef int vsi4 __attribute__((vector_size(16)));
typedef __attribute__((address_space(1))) vsi4 as1_vsi4;
typedef __attribute__((address_space(3))) vsi4 as3_vsi4;

__device__ __forceinline__ void async_copy16(void* lds_dst, const void* gsrc) {
  // per-lane 16B DMA: global -> LDS, no VGPR round trip, tracked by ASYNCcnt
  __builtin_amdgcn_global_load_async_to_lds_b128(
      (as1_vsi4*)(unsigned long long)gsrc,  // generic==as1 address for global
      (as3_vsi4*)lds_dst,                   // generic -> LDS addrspace cast
      0, 0);
}
#endif

static __device__ __forceinline__ v8f wmma_f32(v2f a, v2f b, v8f c) {
  // D = A(16x4) * B(4x16) + C(16x16), fp32, RNE
  return __builtin_amdgcn_wmma_f32_16x16x4_f32(false, a, false, b, (short)0, c,
                                               false, false);
}

// ---------------------------------------------------------------------------
// GEMM: Y[M,N] = (X[M,K] @ W[N,ldw]^T(cols wOff..wOff+K)) [+Yprev] [+bias]
//                [*rowmask[row]] [act: 1+sigmoid]
// Block tile 32(M) x 64(N), BK=32. 256 threads = 8 waves (2x4 of 16x16 tiles).
// M % 32 == 0, N % 64 == 0, K % 32 == 0 (true for all shapes here).
// ---------------------------------------------------------------------------
__global__ __launch_bounds__(256) void gemm_wmma_f32_k(
    const float* __restrict__ X, const float* __restrict__ W,
    const float* __restrict__ bias, const float* __restrict__ rowmask,
    float* __restrict__ Y, int N, int K, int ldw, int wOff,
    int accumulate, int act) {
#if HAVE_ASYNC_LDS
  __shared__ float As[2][32][36];  // stride 36: 16B-aligned, conflict-free
  __shared__ float Bs[2][64][36];
#else
  __shared__ float As[1][32][36];
  __shared__ float Bs[1][64][36];
#endif

  const int tid  = threadIdx.x;
  const int lane = tid & 31;
  const int wave = tid >> 5;
  const int hl   = lane >> 4;   // lane half (0/1)
  const int l15  = lane & 15;
  const int wm   = wave & 1;    // wave tile row (2)
  const int wn   = wave >> 1;   // wave tile col (4)
  const int m0   = blockIdx.y << 5;
  const int n0   = blockIdx.x << 6;

  v8f acc = {0.f, 0.f, 0.f, 0.f, 0.f, 0.f, 0.f, 0.f};

  const int ar = tid >> 3;        // 0..31
  const int ac = (tid & 7) << 2;  // 0..28

#if HAVE_ASYNC_LDS
  const int nk = K >> 5;
  // prologue: prefetch tile 0 into buffer 0 (3 async B128 per thread)
  async_copy16(&As[0][ar][ac], &X[(size_t)(m0 + ar) * K + ac]);
#pragma unroll
  for (int i = 0; i < 2; ++i) {
    int idx = tid + (i << 8);
    int br  = idx >> 3;
    int bc  = (idx & 7) << 2;
    async_copy16(&Bs[0][br][bc], &W[(size_t)(n0 + br) * ldw + (wOff + bc)]);
  }
  for (int t = 0; t < nk; ++t) {
    const int buf = t & 1;
    if (t + 1 < nk) {
      const int k0 = (t + 1) << 5;
      async_copy16(&As[buf ^ 1][ar][ac],
                   &X[(size_t)(m0 + ar) * K + (k0 + ac)]);
#pragma unroll
      for (int i = 0; i < 2; ++i) {
        int idx = tid + (i << 8);
        int br  = idx >> 3;
        int bc  = (idx & 7) << 2;
        async_copy16(&Bs[buf ^ 1][br][bc],
                     &W[(size_t)(n0 + br) * ldw + (wOff + k0 + bc)]);
      }
      // async loads complete in order: <=3 outstanding => tile t resident
      __builtin_amdgcn_s_wait_asynccnt(3);
    } else {
      __builtin_amdgcn_s_wait_asynccnt(0);
    }
    __syncthreads();
#pragma unroll
    for (int kk = 0; kk < 32; kk += 4) {
      v2f a = *(const v2f*)&As[buf][(wm << 4) + l15][kk + (hl << 1)];
      v2f b = *(const v2f*)&Bs[buf][(wn << 4) + l15][kk + (hl << 1)];
      acc = wmma_f32(a, b, acc);
    }
    __syncthreads();  // everyone done reading buf before it is re-prefetched
  }
#else
  for (int k0 = 0; k0 < K; k0 += 32) {
    *(float4*)&As[0][ar][ac] =
        *(const float4*)&X[(size_t)(m0 + ar) * K + (k0 + ac)];
#pragma unroll
    for (int i = 0; i < 2; ++i) {
      int idx = tid + (i << 8);
      int br  = idx >> 3;
      int bc  = (idx & 7) << 2;
      *(float4*)&Bs[0][br][bc] =
          *(const float4*)&W[(size_t)(n0 + br) * ldw + (wOff + k0 + bc)];
    }
    __syncthreads();
#pragma unroll
    for (int kk = 0; kk < 32; kk += 4) {
      v2f a = *(const v2f*)&As[0][(wm << 4) + l15][kk + (hl << 1)];
      v2f b = *(const v2f*)&Bs[0][(wn << 4) + l15][kk + (hl << 1)];
      acc = wmma_f32(a, b, acc);
    }
    __syncthreads();
  }
#endif

#pragma unroll
  for (int r = 0; r < 8; ++r) {
    int grow  = m0 + (wm << 4) + r + (hl << 3);  // C: VGPR r -> M=r (+8 hi half)
    int gcol  = n0 + (wn << 4) + l15;            //    lane   -> N
    size_t yi = (size_t)grow * N + gcol;
    float val = acc[r];
    if (accumulate) val += Y[yi];
    if (bias) val += bias[gcol];
    if (rowmask) val *= rowmask[grow];
    if (act) val = 1.0f + 1.0f / (1.0f + expf(-val));  // 1 + sigmoid (gates)
    Y[yi] = val;
  }
}

// ---------------------------------------------------------------------------
// Attention: one wave per (batch b, head h, 16-query tile).
//   scores[t,s] = (gQ*Q[t,:]) . (gK*K[s,:]) / sqrt(128), -inf where mask==0
//   P = softmax_s(scores);  out[t,n] = sum_s P[t,s] * (gV*V[s,n])
// dh = 128, H = 8. T,S in {128,256}. Q/K/V are head slices (col offset h*128)
// of row-major [B*T, ld*] buffers; V base pointer already includes the value
// section column offset. gate (optional) is [B, 1024], same gate for q/k/v.
// ---------------------------------------------------------------------------
__global__ __launch_bounds__(32) void attn_wmma_f32_k(
    const float* __restrict__ Qp, int ldq,
    const float* __restrict__ Kp, int ldk,
    const float* __restrict__ Vp, int ldv,
    const float* __restrict__ kmask,   // [B,S]
    const float* __restrict__ gate,    // [B,1024] or null
    float* __restrict__ outp, int ldo, // [B*T, 1024]
    int T, int S) {
  __shared__ float Qs[16][132];   // stride 132: float4 aligned, conflict-free
  __shared__ float Ks[16][132];
  __shared__ float Ps[16][260];   // scores strip, up to S=256 cols

  const int lane = threadIdx.x;
  const int hl   = lane >> 4;
  const int l15  = lane & 15;
  const int b    = blockIdx.z;
  const int h    = blockIdx.y;
  const int hc   = h << 7;        // h*128
  const int t0   = blockIdx.x << 4;
  const size_t qrow0 = (size_t)b * T + t0;
  const size_t srow0 = (size_t)b * S;
  const float* gptr  = gate ? (gate + (size_t)b * O_DIM + hc) : (const float*)0;
  const int c4 = lane << 2;

  // stage Q tile (16 x 128); gate folded in when present
#if HAVE_ASYNC_LDS
  if (!gptr) {
#pragma unroll
    for (int i = 0; i < 16; ++i)
      async_copy16(&Qs[i][c4], &Qp[(qrow0 + i) * ldq + hc + c4]);
    __builtin_amdgcn_s_wait_asynccnt(0);
  } else
#endif
  {
#pragma unroll
    for (int i = 0; i < 16; ++i) {
      float4 qv = *(const float4*)&Qp[(qrow0 + i) * ldq + hc + c4];
      if (gptr) {
        qv.x *= gptr[c4 + 0]; qv.y *= gptr[c4 + 1];
        qv.z *= gptr[c4 + 2]; qv.w *= gptr[c4 + 3];
      }
      *(float4*)&Qs[i][c4] = qv;
    }
  }
  __syncthreads();

  const float scale = 0.088388347648318447f;  // 1/sqrt(128)
  const int nst = S >> 4;
  for (int st = 0; st < nst; ++st) {
    // stage K tile (16 x 128)
#if HAVE_ASYNC_LDS
    if (!gptr) {
#pragma unroll
      for (int i = 0; i < 16; ++i)
        async_copy16(&Ks[i][c4],
                     &Kp[(srow0 + (st << 4) + i) * ldk + hc + c4]);
      __builtin_amdgcn_s_wait_asynccnt(0);
    } else
#endif
    {
#pragma unroll
      for (int i = 0; i < 16; ++i) {
        float4 kv =
            *(const float4*)&Kp[(srow0 + (st << 4) + i) * ldk + hc + c4];
        if (gptr) {
          kv.x *= gptr[c4 + 0]; kv.y *= gptr[c4 + 1];
          kv.z *= gptr[c4 + 2]; kv.w *= gptr[c4 + 3];
        }
        *(float4*)&Ks[i][c4] = kv;
      }
    }
    __syncthreads();
    v8f acc = {0.f, 0.f, 0.f, 0.f, 0.f, 0.f, 0.f, 0.f};
#pragma unroll
    for (int j = 0; j < 32; ++j) {  // dh/4 WMMA steps
      v2f a  = *(const v2f*)&Qs[l15][(j << 2) + (hl << 1)];
      v2f bb = *(const v2f*)&Ks[l15][(j << 2) + (hl << 1)];  // B = K^T
      acc = wmma_f32(a, bb, acc);
    }
    int scol = (st << 4) + l15;
    float mk = kmask[srow0 + scol];
#pragma unroll
    for (int r = 0; r < 8; ++r) {
      float sc = acc[r] * scale;
      if (mk == 0.0f) sc = -__builtin_inff();
      Ps[r + (hl << 3)][scol] = sc;   // C layout: row = r + 8*half, col = l15
    }
    __syncthreads();
  }

  // row softmax over S: 2 lanes per row (lane ^ 16 = partner)
  {
    int row  = l15;
    int sBeg = hl * (S >> 1);
    int sEnd = sBeg + (S >> 1);
    float m = -__builtin_inff();
    for (int s = sBeg; s < sEnd; ++s) m = fmaxf(m, Ps[row][s]);
    m = fmaxf(m, __shfl_xor(m, 16, 32));
    float sum = 0.f;
    for (int s = sBeg; s < sEnd; ++s) {
      float e = expf(Ps[row][s] - m);
      Ps[row][s] = e;
      sum += e;
    }
    sum += __shfl_xor(sum, 16, 32);
    float inv = 1.0f / sum;
    for (int s = sBeg; s < sEnd; ++s) Ps[row][s] *= inv;
  }
  __syncthreads();

  // out tile (16 x 128) = P (16 x S) @ V (S x 128), gate folded into V
  const float* Vb = Vp + srow0 * ldv + hc;
  const int nkj = S >> 2;
#pragma unroll
  for (int nt = 0; nt < 8; ++nt) {
    int ncol = (nt << 4) + l15;
    float gv = gptr ? gptr[ncol] : 1.0f;
    v8f acc = {0.f, 0.f, 0.f, 0.f, 0.f, 0.f, 0.f, 0.f};
    for (int j = 0; j < nkj; ++j) {
      v2f a = *(const v2f*)&Ps[l15][(j << 2) + (hl << 1)];
      int srow = (j << 2) + (hl << 1);
      v2f bb;
      bb.x = Vb[(size_t)srow * ldv + ncol] * gv;        // coalesced per half
      bb.y = Vb[(size_t)(srow + 1) * ldv + ncol] * gv;
      acc = wmma_f32(a, bb, acc);
    }
#pragma unroll
    for (int r = 0; r < 8; ++r) {
      outp[(qrow0 + r + (hl << 3)) * ldo + hc + ncol] = acc[r];
    }
  }
}

// ---------------------------------------------------------------------------
// masked mean over tokens: out[b,d] = sum_t X[b,t,d]*mask[b,t] / sum_t mask
// ---------------------------------------------------------------------------
__global__ __launch_bounds__(256) void masked_mean_k(
    const float* __restrict__ X, const float* __restrict__ mask,
    float* __restrict__ out, int T) {
  int b = blockIdx.y;
  int d = (blockIdx.x << 8) + threadIdx.x;
  const float* xb = X + (size_t)b * T * O_DIM + d;
  const float* mb = mask + (size_t)b * T;
  float s = 0.f, ms = 0.f;
  for (int t = 0; t < T; ++t) {
    float mk = mb[t];
    s += xb[(size_t)t * O_DIM] * mk;
    ms += mk;
  }
  out[(size_t)b * O_DIM + d] = s / ms;
}

// ---------------------------------------------------------------------------
// elementwise helpers (float4 granularity; all sizes are multiples of 4)
// ---------------------------------------------------------------------------
__global__ void ew_copy_k(const float* __restrict__ a, float* __restrict__ o,
                          int n4) {
  int i = blockIdx.x * blockDim.x + threadIdx.x;
  if (i < n4) ((float4*)o)[i] = ((const float4*)a)[i];
}
__global__ void ew_acc_k(const float* __restrict__ a, float* __restrict__ o,
                         int n4) {
  int i = blockIdx.x * blockDim.x + threadIdx.x;
  if (i < n4) {
    float4 x = ((const float4*)a)[i];
    float4 y = ((float4*)o)[i];
    y.x += x.x; y.y += x.y; y.z += x.z; y.w += x.w;
    ((float4*)o)[i] = y;
  }
}
__global__ void ew_add2_k(const float* __restrict__ a,
                          const float* __restrict__ b, float* __restrict__ o,
                          int n4) {
  int i = blockIdx.x * blockDim.x + threadIdx.x;
  if (i < n4) {
    float4 x = ((const float4*)a)[i];
    float4 y = ((const float4*)b)[i];
    x.x += y.x; x.y += y.y; x.z += y.z; x.w += y.w;
    ((float4*)o)[i] = x;
  }
}
__global__ void ew_add3_k(const float* __restrict__ a,
                          const float* __restrict__ b,
                          const float* __restrict__ c, float* __restrict__ o,
                          int n4) {
  int i = blockIdx.x * blockDim.x + threadIdx.x;
  if (i < n4) {
    float4 x = ((const float4*)a)[i];
    float4 y = ((const float4*)b)[i];
    float4 z = ((const float4*)c)[i];
    x.x += y.x + z.x; x.y += y.y + z.y; x.z += y.z + z.z; x.w += y.w + z.w;
    ((float4*)o)[i] = x;
  }
}

// ---------------------------------------------------------------------------
extern "C" void kernel_launch(void* const* d_in, const int* in_sizes, int n_in,
                              void* d_out, int out_size, void* d_ws,
                              size_t ws_size, hipStream_t stream) {
  (void)in_sizes; (void)n_in; (void)out_size;

  const float* v_in  = (const float*)d_in[0];
  const float* q_in  = (const float*)d_in[1];
  const float* vmask = (const float*)d_in[2];
  const float* qmask = (const float*)d_in[3];
  const float* W_vlin = (const float*)d_in[4];
  const float* b_vlin = (const float*)d_in[5];
  const float* W_qlin = (const float*)d_in[6];
  const float* b_qlin = (const float*)d_in[7];
  const float* W_v2q_src = (const float*)d_in[8];
  const float* b_v2q_src = (const float*)d_in[9];
  const float* W_v2q_tgt = (const float*)d_in[10];
  const float* b_v2q_tgt = (const float*)d_in[11];
  const float* W_v2q_out = (const float*)d_in[12];
  const float* b_v2q_out = (const float*)d_in[13];
  const float* W_q2v_src = (const float*)d_in[14];
  const float* b_q2v_src = (const float*)d_in[15];
  const float* W_q2v_tgt = (const float*)d_in[16];
  const float* b_q2v_tgt = (const float*)d_in[17];
  const float* W_q2v_out = (const float*)d_in[18];
  const float* b_q2v_out = (const float*)d_in[19];
  const float* W_v4q = (const float*)d_in[20];
  const float* b_v4q = (const float*)d_in[21];
  const float* W_q4v = (const float*)d_in[22];
  const float* b_q4v = (const float*)d_in[23];
  const float* W_iv  = (const float*)d_in[24];
  const float* b_iv  = (const float*)d_in[25];
  const float* W_iq  = (const float*)d_in[26];
  const float* b_iq  = (const float*)d_in[27];
  const float* W_vout = (const float*)d_in[28];
  const float* b_vout = (const float*)d_in[29];
  const float* W_qout = (const float*)d_in[30];
  const float* b_qout = (const float*)d_in[31];

  const int MV = 32 * 256;  // 8192 v tokens
  const int MQ = 32 * 128;  // 4096 q tokens
  const int nv4 = MV * O_DIM / 4;
  const int nq4 = MQ * O_DIM / 4;

  // workspace layout (fp32)
  float* ws = (float*)d_ws;
  size_t off = 0;
  const size_t need = (size_t)MV * 3072 + (size_t)MQ * 3072 +
                      5 * (size_t)MV * O_DIM + 5 * (size_t)MQ * O_DIM +
                      4 * (size_t)32 * O_DIM;
  if (ws_size < need * sizeof(float)) return;  // workspace too small

  float* VT  = ws + off; off += (size_t)MV * 3072;   // v-side trans scratch
  float* QT  = ws + off; off += (size_t)MQ * 3072;   // q-side trans scratch
  float* Vin = ws + off; off += (size_t)MV * O_DIM;  // v_c[-1]
  float* V1  = ws + off; off += (size_t)MV * O_DIM;
  float* V2  = ws + off; off += (size_t)MV * O_DIM;
  float* VX  = ws + off; off += (size_t)MV * O_DIM;
  float* VU  = ws + off; off += (size_t)MV * O_DIM;
  float* Qin = ws + off; off += (size_t)MQ * O_DIM;
  float* Q1  = ws + off; off += (size_t)MQ * O_DIM;
  float* Q2  = ws + off; off += (size_t)MQ * O_DIM;
  float* QX  = ws + off; off += (size_t)MQ * O_DIM;
  float* QU  = ws + off; off += (size_t)MQ * O_DIM;
  float* vmean = ws + off; off += (size_t)32 * O_DIM;
  float* qmean = ws + off; off += (size_t)32 * O_DIM;
  float* Gv = ws + off; off += (size_t)32 * O_DIM;
  float* Gq = ws + off; off += (size_t)32 * O_DIM;

  float* RV = (float*)d_out;                 // sum(res_v) [32,256,1024]
  float* RQ = RV + (size_t)MV * O_DIM;       // sum(res_q) [32,128,1024]

  auto gemm = [&](const float* X, const float* W, const float* bias,
                  const float* rmask, float* Y, int M, int N, int K, int ldw,
                  int wOff, int accF, int act) {
    dim3 g(N / 64, M / 32);
    gemm_wmma_f32_k<<<g, dim3(256), 0, stream>>>(X, W, bias, rmask, Y, N, K,
                                                 ldw, wOff, accF, act);
  };
  auto attn = [&](const float* Q, int ldq, const float* K, int ldk,
                  const float* V, int ldv, const float* km, const float* gate,
                  float* out, int T, int S) {
    dim3 g(T / 16, 8, 32);
    attn_wmma_f32_k<<<g, dim3(32), 0, stream>>>(Q, ldq, K, ldk, V, ldv, km,
                                                gate, out, O_DIM, T, S);
  };
  auto copy = [&](const float* a, float* o, int n4) {
    ew_copy_k<<<(n4 + 255) / 256, 256, 0, stream>>>(a, o, n4);
  };
  auto accum = [&](const float* a, float* o, int n4) {
    ew_acc_k<<<(n4 + 255) / 256, 256, 0, stream>>>(a, o, n4);
  };
  auto add2 = [&](const float* a, const float* b, float* o, int n4) {
    ew_add2_k<<<(n4 + 255) / 256, 256, 0, stream>>>(a, b, o, n4);
  };
  auto add3 = [&](const float* a, const float* b, const float* c, float* o,
                  int n4) {
    ew_add3_k<<<(n4 + 255) / 256, 256, 0, stream>>>(a, b, c, o, n4);
  };

  // ---- input projections: v0, q0 ----
  gemm(v_in, W_vlin, b_vlin, nullptr, Vin, MV, 1024, 2048, 2048, 0, 0, 0);
  gemm(q_in, W_qlin, b_qlin, nullptr, Qin, MQ, 1024, 1024, 1024, 0, 0, 0);
  copy(Vin, RV, nv4);   // res_v starts at v0
  copy(Qin, RQ, nq4);   // res_q starts at q0

  for (int blk = 0; blk < 2; ++blk) {
    // ---- inter v->q : q1 = out([Qin, attn(Qin_t, Vin_kv)]) ----
    gemm(Vin, W_v2q_src, b_v2q_src, vmask, VT, MV, 2048, 1024, 1024, 0, 0, 0);
    gemm(Qin, W_v2q_tgt, b_v2q_tgt, qmask, QT, MQ, 1024, 1024, 1024, 0, 0, 0);
    attn(QT, 1024, VT, 2048, VT + 1024, 2048, vmask, nullptr, QU, 128, 256);
    gemm(Qin, W_v2q_out, nullptr, nullptr, Q1, MQ, 1024, 1024, 2048, 0, 0, 0);
    gemm(QU, W_v2q_out, b_v2q_out, nullptr, Q1, MQ, 1024, 1024, 2048, 1024, 1, 0);
    accum(Q1, RQ, nq4);
    add2(Q1, Qin, QX, nq4);  // QX = q1 + q_prev (src for q2v, input for intra)

    // ---- inter q->v : v1 = out([Vin, attn(Vin_t, QX_kv)]) ----
    gemm(QX, W_q2v_src, b_q2v_src, qmask, QT, MQ, 2048, 1024, 1024, 0, 0, 0);
    gemm(Vin, W_q2v_tgt, b_q2v_tgt, vmask, VT, MV, 1024, 1024, 1024, 0, 0, 0);
    attn(VT, 1024, QT, 2048, QT + 1024, 2048, qmask, nullptr, VU, 256, 128);
    gemm(Vin, W_q2v_out, nullptr, nullptr, V1, MV, 1024, 1024, 2048, 0, 0, 0);
    gemm(VU, W_q2v_out, b_q2v_out, nullptr, V1, MV, 1024, 1024, 2048, 1024, 1, 0);
    accum(V1, RV, nv4);
    add2(V1, Vin, VX, nv4);  // VX = v1 + v_prev (input for intra)

    // ---- intra (dynamic gated self-attention) ----
    masked_mean_k<<<dim3(4, 32), 256, 0, stream>>>(VX, vmask, vmean, 256);
    masked_mean_k<<<dim3(4, 32), 256, 0, stream>>>(QX, qmask, qmean, 128);
    gemm(vmean, W_v4q, b_v4q, nullptr, Gq, 32, 1024, 1024, 1024, 0, 0, 1); // 1+sigmoid
    gemm(qmean, W_q4v, b_q4v, nullptr, Gv, 32, 1024, 1024, 1024, 0, 0, 1);
    gemm(VX, W_iv, b_iv, vmask, VT, MV, 3072, 1024, 1024, 0, 0, 0);  // key|qry|val
    gemm(QX, W_iq, b_iq, qmask, QT, MQ, 3072, 1024, 1024, 0, 0, 0);
    attn(VT + 1024, 3072, VT, 3072, VT + 2048, 3072, vmask, Gv, VU, 256, 256);
    attn(QT + 1024, 3072, QT, 3072, QT + 2048, 3072, qmask, Gq, QU, 128, 128);
    add2(VX, VU, VX, nv4);   // v + v_upd
    add2(QX, QU, QX, nq4);   // q + q_upd
    gemm(VX, W_vout, b_vout, nullptr, V2, MV, 1024, 1024, 1024, 0, 0, 0);
    gemm(QX, W_qout, b_qout, nullptr, Q2, MQ, 1024, 1024, 1024, 0, 0, 0);
    accum(V2, RV, nv4);
    accum(Q2, RQ, nq4);

    // next block inputs: v_next = v2 + v1 + v_prev (in place into Vin/Qin)
    add3(V2, V1, Vin, Vin, nv4);
    add3(Q2, Q1, Qin, Qin, nq4);
  }
}